// local_planar_guidance_86921548136559
// MI455X (gfx1250) — compile-verified
//
#include <hip/hip_runtime.h>

#define B_  8
#define HC  88
#define WC  288
#define K   4
#define H_  (HC*K)          // 352
#define W_  (WC*K)          // 1152
#define P_  9
#define NTHREADS WC         // 288 threads = 9 wave32; one thread per column-group
#define LDS_ELEMS (3*4*WC)  // 3456 floats = 13.8 KB

typedef float v4f __attribute__((ext_vector_type(4)));

#if defined(__gfx1250__) && __has_builtin(__builtin_amdgcn_global_load_async_to_lds_b32)
#define LPG_ASYNC 1
typedef __attribute__((address_space(1))) int* gptr_i32;
typedef __attribute__((address_space(3))) int* lptr_i32;
#else
#define LPG_ASYNC 0
#endif

// rcp + 1 Newton-Raphson step: ~0.5-1 ulp, 4 VALU ops vs ~10 for exact '/'
__device__ __forceinline__ float fast_div(float num, float den)
{
#if __has_builtin(__builtin_amdgcn_rcpf)
    float r = __builtin_amdgcn_rcpf(den);
    r = __builtin_fmaf(__builtin_fmaf(-den, r, 1.0f), r, r);
    return num * r;
#else
    return num / den;
#endif
}

__global__ __launch_bounds__(NTHREADS)
void lpg_kernel(const float* __restrict__ pe, float* __restrict__ out)
{
    __shared__ float lds[LDS_ELEMS];

    const int h   = blockIdx.x;   // 0..351
    const int b   = blockIdx.y;   // 0..7
    const int tid = threadIdx.x;  // 0..287

    // the three coarse rows this output row needs (h_shift = -4, 0, +4)
    int row[3];
#pragma unroll
    for (int s = 0; s < 3; ++s) {
        int ih = h + (s - 1) * K;
        ih = ih < 0 ? 0 : (ih > H_ - 1 ? H_ - 1 : ih);
        row[s] = ih >> 2;
    }

    // Stage 3 coarse rows x 4 channels x 288 cols into LDS.
    // e = it*288 + tid  =>  slot = it>>2, ch = it&3, c = tid  (no divides).
    // CDNA5 path: async global->LDS DMA, tracked by ASYNCcnt.
    const float* peb = pe + (size_t)b * 4 * HC * WC + tid;
#pragma unroll
    for (int it = 0; it < 12; ++it) {
        const int slot = it >> 2;
        const int ch   = it & 3;
        const float* src = peb + (ch * HC + row[slot]) * WC;
#if LPG_ASYNC
        __builtin_amdgcn_global_load_async_to_lds_b32(
            (gptr_i32)src, (lptr_i32)(&lds[it * WC + tid]), 0, 0);
#else
        lds[it * WC + tid] = *src;
#endif
    }
#if LPG_ASYNC
    asm volatile("s_wait_asynccnt 0" ::: "memory");
#endif
    __syncthreads();

    const float vv = ((float)(h & (K - 1)) - 1.5f) * 0.25f;  // v[h]

    // per-plane coarse columns: sw = -4 -> tid-1, 0 -> tid, +4 -> tid+1 (clamped)
    const int cm = tid > 0        ? tid - 1 : 0;
    const int c0 = tid;
    const int cp = tid < (WC - 1) ? tid + 1 : WC - 1;

    // 32-bit element offset (max 29.2M < 2^31) -> saddr-form global stores
    const unsigned int obase =
        ((unsigned int)(b * P_) * H_ + (unsigned int)h) * W_ + (unsigned int)(tid * 4);

#pragma unroll
    for (int p = 0; p < P_; ++p) {
        const int i  = p / 3;          // h_shift index   (constant-folded)
        const int j  = p - i * 3;      // w_shift index
        const int sh = (i - 1) * K;
        const int sw = (j - 1) * K;
        const int c  = (j == 0) ? cm : (j == 1) ? c0 : cp;

        const float n0 = lds[(i * 4 + 0) * WC + c];
        const float n1 = lds[(i * 4 + 1) * WC + c];
        const float n2 = lds[(i * 4 + 2) * WC + c];
        const float n3 = lds[(i * 4 + 3) * WC + c];

        // den_t = n0*(u_t - sh) + n1*(vv - sw) + n2 ;  u_t = (t-1.5)/4
        const float C = n1 * (vv - (float)sw) + n2 - n0 * (float)sh;

        v4f r;
        r.x = fast_div(n3, __builtin_fmaf(n0, -0.375f, C));
        r.y = fast_div(n3, __builtin_fmaf(n0, -0.125f, C));
        r.z = fast_div(n3, __builtin_fmaf(n0,  0.125f, C));
        r.w = fast_div(n3, __builtin_fmaf(n0,  0.375f, C));

        __builtin_nontemporal_store(
            r, (v4f*)(out + obase + (unsigned int)(p * H_ * W_)));
    }
}

extern "C" void kernel_launch(void* const* d_in, const int* in_sizes, int n_in,
                              void* d_out, int out_size, void* d_ws, size_t ws_size,
                              hipStream_t stream) {
    (void)in_sizes; (void)n_in; (void)out_size; (void)d_ws; (void)ws_size;
    const float* pe  = (const float*)d_in[0];
    float*       out = (float*)d_out;
    dim3 grid(H_, B_);    // one block per (output row, batch)
    dim3 block(NTHREADS); // 288 = 9 wave32
    lpg_kernel<<<grid, block, 0, stream>>>(pe, out);
}